// LinearDynamicalSystem_34711925686725
// MI455X (gfx1250) — compile-verified
//
#include <hip/hip_runtime.h>

// Batched 2nd-order IIR filter (direct-form II transposed) for MI455X.
//
//   y_t = b0*u_t + z0_{t-1}
//   z0_t = b1*u_t - a1*y_t + z1_{t-1}
//   z1_t = b2*u_t - a2*y_t            (b2 == 0 in the reference)
//
// Strategy (3 passes):
//   1) seg_state_kernel:  per (batch, 256-step segment), particular exit state
//      assuming zero entry state (scalar recurrence, B*NSEG threads).
//   2) scan_state_kernel: per batch, scan segment states:
//      z_in[s+1] = M^256 * z_in[s] + z_part[s],   M = [[-a1,1],[-a2,0]]
//   3) wmma_filter_kernel: per (segment, 16-batch group) wave, 16 chunks of 16
//      timesteps. Each chunk:  Y(16t x 16b) = H(16x16) * U(16x16) + homog(z_in)
//      computed with 4x V_WMMA_F32_16X16X4_F32, state carried via shuffles.
//      H and the homogeneous coefficients live in LDS (built once per block),
//      so the steady-state loop is loads -> 4x wmma -> NT stores -> shuffles.
//
// Cache policy: u_in (64 MiB) is read by pass 1 and re-read by pass 3; with
// 192 MB L2 the second read hits L2 if we don't pollute it, so y stores are
// non-temporal. Effective HBM traffic ~128 MiB -> ~5.5 us at 23.3 TB/s.

typedef __attribute__((ext_vector_type(2))) float v2f;
typedef __attribute__((ext_vector_type(4))) float v4f;
typedef __attribute__((ext_vector_type(8))) float v8f;

#define SEG   256
#define CHUNK 16
#define CHUNKS_PER_SEG (SEG / CHUNK)

// ---------------- Pass 1: particular exit state per segment ----------------
__global__ void seg_state_kernel(const float* __restrict__ u,
                                 const float* __restrict__ bc,
                                 const float* __restrict__ fc,
                                 float* __restrict__ zpart,
                                 int Bn, int T, int nseg) {
  int tid = blockIdx.x * blockDim.x + threadIdx.x;
  int total = Bn * nseg;
  if (tid >= total) return;
  int b = tid % Bn;
  int s = tid / Bn;
  const float b0 = bc[0], b1 = bc[1], b2v = 0.0f;
  const float a1 = fc[0], a2 = fc[1];
  float z0 = 0.f, z1 = 0.f;
  const float* up = u + (long long)b * T + (long long)s * SEG;
  #pragma unroll 4
  for (int t = 0; t < SEG; ++t) {
    float ut = up[t];
    float yv  = b0 * ut + z0;
    float nz0 = b1 * ut - a1 * yv + z1;
    float nz1 = b2v * ut - a2 * yv;
    z0 = nz0; z1 = nz1;
  }
  zpart[(size_t)(s * Bn + b) * 2 + 0] = z0;
  zpart[(size_t)(s * Bn + b) * 2 + 1] = z1;
}

// ---------------- Pass 2: scan segment entry states per batch --------------
__global__ void scan_state_kernel(const float* __restrict__ bc,
                                  const float* __restrict__ fc,
                                  const float* __restrict__ yinit,
                                  const float* __restrict__ uinit,
                                  const float* __restrict__ zpart,
                                  float* __restrict__ zin,
                                  int Bn, int nseg) {
  int b = blockIdx.x * blockDim.x + threadIdx.x;
  if (b >= Bn) return;
  const float b1 = bc[1];
  const float a1 = fc[0], a2 = fc[1];

  // M^SEG by repeated squaring (SEG = 256 = 2^8), M = [[-a1, 1], [-a2, 0]]
  float m00 = -a1, m01 = 1.f, m10 = -a2, m11 = 0.f;
  #pragma unroll
  for (int i = 0; i < 8; ++i) {
    float t00 = m00 * m00 + m01 * m10;
    float t01 = m00 * m01 + m01 * m11;
    float t10 = m10 * m00 + m11 * m10;
    float t11 = m10 * m01 + m11 * m11;
    m00 = t00; m01 = t01; m10 = t10; m11 = t11;
  }

  // lfiltic (matches reference exactly)
  float z0 = b1 * uinit[b * 2 + 0] - a1 * yinit[b * 2 + 0] - a2 * yinit[b * 2 + 1];
  float z1 = -a2 * yinit[b * 2 + 0];

  for (int s = 0; s < nseg; ++s) {
    size_t idx = (size_t)(s * Bn + b) * 2;
    zin[idx + 0] = z0;
    zin[idx + 1] = z1;
    float nz0 = m00 * z0 + m01 * z1 + zpart[idx + 0];
    float nz1 = m10 * z0 + m11 * z1 + zpart[idx + 1];
    z0 = nz0; z1 = nz1;
  }
}

// ---------------- Pass 3: WMMA chunked filter ------------------------------
__global__ __launch_bounds__(256)
void wmma_filter_kernel(const float* __restrict__ u,
                        const float* __restrict__ bc,
                        const float* __restrict__ fc,
                        const float* __restrict__ zin,
                        float* __restrict__ y,
                        int Bn, int T, int nseg) {
  // H[t][k] = h[t-k] (t>=k else 0), lower-triangular Toeplitz impulse response.
  // Cs[k] = (c0_k, c1_k): homogeneous response row coefficients.
  __shared__ float Hs[16][16];
  __shared__ v2f   Cs[16];

  const float b0 = bc[0], b1 = bc[1], b2v = 0.0f;
  const float a1 = fc[0], a2 = fc[1];

  // --- build coefficient tables once per block (no per-lane select chains) -
  if (threadIdx.x < 16) {
    const int t = (int)threadIdx.x;
    float hm1 = 0.f, hm2 = 0.f;
    #pragma unroll
    for (int d = 0; d < 16; ++d) {
      float hd;
      if (d == 0)      hd = b0;
      else if (d == 1) hd = b1 - a1 * b0;
      else if (d == 2) hd = b2v - a1 * hm1 - a2 * hm2;
      else             hd = -a1 * hm1 - a2 * hm2;
      if (d <= t) Hs[t][t - d] = hd;   // fill diagonal d of row t
      hm2 = hm1; hm1 = hd;
    }
    for (int k = t + 1; k < 16; ++k) Hs[t][k] = 0.f;  // upper triangle
  } else if (threadIdx.x == 16) {
    float cc0 = 1.f, cc1 = 0.f;
    #pragma unroll
    for (int k = 0; k < 16; ++k) {
      v2f c; c.x = cc0; c.y = cc1;
      Cs[k] = c;
      float n0 = -a1 * cc0 - a2 * cc1;
      cc1 = cc0; cc0 = n0;
    }
  }
  __syncthreads();

  const int lane = (int)(threadIdx.x & 31);
  const int wave = (int)((blockIdx.x * blockDim.x + threadIdx.x) >> 5);
  const int ngroups = Bn >> 4;
  const int totalWaves = nseg * ngroups;

  if (wave < totalWaves) {             // whole-wave uniform: EXEC full inside
    const int seg  = wave / ngroups;
    const int grp  = wave % ngroups;
    const int col  = lane & 15;        // N column == batch-in-group; A row
    const int half = lane >> 4;        // K-pair / C-row-half selector
    const int b    = (grp << 4) + col;

    // loop-invariant fragments from LDS (ds_load_b64)
    v2f afrag[4];
    #pragma unroll
    for (int j = 0; j < 4; ++j)
      afrag[j] = *(const v2f*)&Hs[col][4 * j + 2 * half];

    v2f crow[8];
    #pragma unroll
    for (int r = 0; r < 8; ++r)
      crow[r] = Cs[r + 8 * half];

    // entry state for this segment & this lane's batch (duplicated per half)
    float z0 = zin[(size_t)(seg * Bn + b) * 2 + 0];
    float z1 = zin[(size_t)(seg * Bn + b) * 2 + 1];

    const long long rowbase = (long long)b * T + (long long)seg * SEG;

    for (int c = 0; c < CHUNKS_PER_SEG; ++c) {
      const long long t0 = rowbase + (long long)c * CHUNK;

      // prefetch next chunk of this batch row (global_prefetch_b8)
      if (c + 1 < CHUNKS_PER_SEG)
        __builtin_prefetch(u + t0 + CHUNK, 0, 1);

      // B fragments: U[4j + 2*half + {0,1}][col], float2 global loads
      v2f bfrag[4];
      #pragma unroll
      for (int j = 0; j < 4; ++j)
        bfrag[j] = *(const v2f*)(u + t0 + 4 * j + 2 * half);

      // accumulator seeded with homogeneous response of entry state
      v8f acc;
      #pragma unroll
      for (int r = 0; r < 8; ++r) acc[r] = crow[r].x * z0 + crow[r].y * z1;

      // Y += H * U : four v_wmma_f32_16x16x4_f32 over the K=16 dimension
      acc = __builtin_amdgcn_wmma_f32_16x16x4_f32(false, afrag[0], false, bfrag[0], (short)0, acc, false, false);
      acc = __builtin_amdgcn_wmma_f32_16x16x4_f32(false, afrag[1], false, bfrag[1], (short)0, acc, false, false);
      acc = __builtin_amdgcn_wmma_f32_16x16x4_f32(false, afrag[2], false, bfrag[2], (short)0, acc, false, false);
      acc = __builtin_amdgcn_wmma_f32_16x16x4_f32(false, afrag[3], false, bfrag[3], (short)0, acc, false, false);

      // store: lane's 8 outputs are times t0 + 8*half + 0..7 (contiguous).
      // Non-temporal: y is never re-read; keep L2 free to retain u_in.
      v4f lo; lo.x = acc[0]; lo.y = acc[1]; lo.z = acc[2]; lo.w = acc[3];
      v4f hi; hi.x = acc[4]; hi.y = acc[5]; hi.z = acc[6]; hi.w = acc[7];
      __builtin_nontemporal_store(lo, (v4f*)(y + t0 + 8 * half + 0));
      __builtin_nontemporal_store(hi, (v4f*)(y + t0 + 8 * half + 4));

      // exit state: needs y14, y15, u14, u15 (live in upper-half lane)
      float y14 = __shfl(acc[6],     col + 16, 32);
      float y15 = __shfl(acc[7],     col + 16, 32);
      float u14 = __shfl(bfrag[3].x, col + 16, 32);
      float u15 = __shfl(bfrag[3].y, col + 16, 32);
      float nz1 = b2v * u15 - a2 * y15;
      float nz0 = b1 * u15 - a1 * y15 + b2v * u14 - a2 * y14;
      z0 = nz0; z1 = nz1;
    }
  }
}

// ---------------- Fallback: naive one-thread-per-batch scan ----------------
__global__ void naive_iir_kernel(const float* __restrict__ u,
                                 const float* __restrict__ bc,
                                 const float* __restrict__ fc,
                                 const float* __restrict__ yinit,
                                 const float* __restrict__ uinit,
                                 float* __restrict__ y, int Bn, int T) {
  int b = blockIdx.x * blockDim.x + threadIdx.x;
  if (b >= Bn) return;
  const float b0 = bc[0], b1 = bc[1];
  const float a1 = fc[0], a2 = fc[1];
  float z0 = b1 * uinit[b * 2 + 0] - a1 * yinit[b * 2 + 0] - a2 * yinit[b * 2 + 1];
  float z1 = -a2 * yinit[b * 2 + 0];
  const float* up = u + (long long)b * T;
  float* yp = y + (long long)b * T;
  for (int t = 0; t < T; ++t) {
    float ut = up[t];
    float yv = b0 * ut + z0;
    float nz0 = b1 * ut - a1 * yv + z1;
    z1 = -a2 * yv;
    z0 = nz0;
    yp[t] = yv;
  }
}

extern "C" void kernel_launch(void* const* d_in, const int* in_sizes, int n_in,
                              void* d_out, int out_size, void* d_ws, size_t ws_size,
                              hipStream_t stream) {
  const float* bc = (const float*)d_in[0];   // b_coeff (2)
  const float* fc = (const float*)d_in[1];   // f_coeff (2)
  const float* u  = (const float*)d_in[2];   // u_in (B*T)
  const float* yi = (const float*)d_in[3];   // y_init (B*2)
  const float* ui = (const float*)d_in[4];   // u_init (B*2)
  float* y = (float*)d_out;

  const int Bn = in_sizes[3] / 2;
  const int T  = in_sizes[2] / Bn;
  const int nseg = T / SEG;
  const size_t need = (size_t)2 * (size_t)nseg * (size_t)Bn * 2 * sizeof(float);

  if ((T % SEG) != 0 || (Bn % 16) != 0 || ws_size < need) {
    naive_iir_kernel<<<(Bn + 255) / 256, 256, 0, stream>>>(u, bc, fc, yi, ui, y, Bn, T);
    return;
  }

  float* zpart = (float*)d_ws;                         // [nseg][B][2]
  float* zin   = zpart + (size_t)nseg * Bn * 2;        // [nseg][B][2]

  const int total = Bn * nseg;
  seg_state_kernel<<<(total + 255) / 256, 256, 0, stream>>>(u, bc, fc, zpart, Bn, T, nseg);
  scan_state_kernel<<<(Bn + 255) / 256, 256, 0, stream>>>(bc, fc, yi, ui, zpart, zin, Bn, nseg);

  const int waves  = nseg * (Bn / 16);
  const int blocks = (waves + 7) / 8;                  // 256 threads = 8 waves/block
  wmma_filter_kernel<<<blocks, 256, 0, stream>>>(u, bc, fc, zin, y, Bn, T, nseg);
}